// PoNQ_59880434040938
// MI455X (gfx1250) — compile-verified
//
#include <hip/hip_runtime.h>
#include <hip/hip_bf16.h>
#include <math.h>

typedef float v2f __attribute__((ext_vector_type(2)));
typedef float v8f __attribute__((ext_vector_type(8)));
typedef unsigned int u32x4 __attribute__((ext_vector_type(4)));
typedef int i32x4 __attribute__((ext_vector_type(4)));
typedef int i32x8 __attribute__((ext_vector_type(8)));

#define NPTS   2048
#define NSAMP  131072
#define EPS_R  0.05f

#if defined(__has_builtin)
#if __has_builtin(__builtin_amdgcn_tensor_load_to_lds) && __has_builtin(__builtin_amdgcn_s_wait_tensorcnt)
#define HAVE_TDM 1
#endif
#endif
#ifndef HAVE_TDM
#define HAVE_TDM 0
#endif

// ---------------------------------------------------------------------------
// Kernel 0: zero the output buffer (Q / nsum / counts are accumulated there)
// ---------------------------------------------------------------------------
__global__ void ponq_zero(float* __restrict__ out, int n) {
    int i = blockIdx.x * blockDim.x + threadIdx.x;
    if (i < n) out[i] = 0.0f;
}

// ---------------------------------------------------------------------------
// Kernel 1: nearest-point assignment via V_WMMA_F32_16X16X4_F32.
// score[m][n] = |p_n|^2 - 2 s_m . p_n  ==  A(16x4) x B(4x16)
//   A row m  = [sx, sy, sz, 1]
//   B col n  = [-2px, -2py, -2pz, |p|^2]
// Points staged into LDS via TDM (tensor_load_to_lds) when available.
// Rolled loop (unroll 2) — occupancy hides LDS latency; no manual pipelining
// (full unroll caused catastrophic VGPR spill in a previous revision).
// ---------------------------------------------------------------------------
__global__ __launch_bounds__(256) void ponq_closest(
    const float* __restrict__ points,
    const float* __restrict__ samples,
    int* __restrict__ closest)
{
#if HAVE_TDM
    __shared__ float pts_lds[NPTS * 3];   // 24 KB raw points (TDM destination)
#endif
    __shared__ float bmat[NPTS * 4];      // 32 KB B-matrix staging

    const int tid  = threadIdx.x;
    const int lane = tid & 31;
    const int wave = tid >> 5;
    const int s0   = (blockIdx.x * 8 + wave) * 16;  // 16 samples per wave
    const int m    = lane & 15;                     // row / col within 16-tile
    const int hi   = lane >> 4;                     // half: selects K pair + row block

    // Prefetch this wave's sample rows while LDS fills (global_prefetch_b8).
    __builtin_prefetch(samples + 3 * (s0 + m), 0, 1);

#if HAVE_TDM
    if (wave == 0) {
        // ---- Tensor DMA descriptor: 1D tile, 6144 x f32, global -> LDS ----
        unsigned long long ga = (unsigned long long)(uintptr_t)points;
        unsigned lds_base = (unsigned)(uintptr_t)&pts_lds[0];
        u32x4 g0;
        g0[0] = 1u;                                        // count=1 (valid D#)
        g0[1] = lds_base;                                  // lds_addr
        g0[2] = (unsigned)(ga & 0xFFFFFFFFu);              // global_addr[31:0]
        g0[3] = (unsigned)((ga >> 32) & 0x1FFFFFFull)      // global_addr[56:32]
              | (2u << 30);                                // type=2 ("image")
        i32x8 g1;
        g1[0] = (int)(2u << 16);        // workgroup_mask=0, data_size=2 (4B)
        g1[1] = (int)(6144u << 16);     // tensor_dim0[15:0] in bits[63:48]
        g1[2] = 0;                      // tensor_dim0[31:16]=0, tensor_dim1 lo=0
        g1[3] = (int)(6144u << 16);     // tile_dim0 = 6144 in bits[127:112]
        g1[4] = 0;                      // tile_dim1/2 unused (1D)
        g1[5] = 6144;                   // tensor_dim0_stride[31:0]
        g1[6] = 0;
        g1[7] = 0;
        i32x4 z4 = {};
#if __clang_major__ >= 23
        i32x8 z8 = {};
        __builtin_amdgcn_tensor_load_to_lds(g0, g1, z4, z4, z8, 0);
#else
        __builtin_amdgcn_tensor_load_to_lds(g0, g1, z4, z4, 0);
#endif
        __builtin_amdgcn_s_wait_tensorcnt(0);
    }
    __syncthreads();
    // Build B-matrix from the LDS copy of the points.
    for (int p = tid; p < NPTS; p += 256) {
        float px = pts_lds[3 * p + 0];
        float py = pts_lds[3 * p + 1];
        float pz = pts_lds[3 * p + 2];
        bmat[4 * p + 0] = -2.0f * px;
        bmat[4 * p + 1] = -2.0f * py;
        bmat[4 * p + 2] = -2.0f * pz;
        bmat[4 * p + 3] = px * px + py * py + pz * pz;
    }
#else
    for (int p = tid; p < NPTS; p += 256) {
        float px = points[3 * p + 0];
        float py = points[3 * p + 1];
        float pz = points[3 * p + 2];
        bmat[4 * p + 0] = -2.0f * px;
        bmat[4 * p + 1] = -2.0f * py;
        bmat[4 * p + 2] = -2.0f * pz;
        bmat[4 * p + 3] = px * px + py * py + pz * pz;
    }
#endif
    __syncthreads();

    // A fragment (16x4 f32): lanes 0-15 hold K=0,1 ; lanes 16-31 hold K=2,3.
    const float* sp = samples + 3 * (s0 + m);
    float sx = sp[0], sy = sp[1], sz = sp[2];
    v2f a;
    a.x = hi ? sz   : sx;
    a.y = hi ? 1.0f : sy;

    float minv[8];
    int   mini[8];
#pragma unroll
    for (int i = 0; i < 8; ++i) { minv[i] = 3.4e38f; mini[i] = 2147483647; }

    const int bidx = 4 * m + (hi << 1);   // float index of this lane's K pair

    // One WMMA + 8 cmp/cndmask-pairs per 16x16 tile. Per-lane scan order
    // visits pi strictly increasing, so plain '<' keeps the first (lowest
    // index) minimum within the lane — no eq-tiebreak needed here.
#pragma unroll 2
    for (int t = 0; t < NPTS / 16; ++t) {
        const int pi = t * 16 + m;   // point (column) handled by this lane
        v2f b = *(const v2f*)&bmat[64 * t + bidx];   // ds_load_b64, 8B aligned
        v8f c = {};
        c = __builtin_amdgcn_wmma_f32_16x16x4_f32(
                /*neg_a=*/false, a, /*neg_b=*/false, b,
                /*c_mod=*/(short)0, c, /*reuse_a=*/false, /*reuse_b=*/false);
        // c[i] = score for row M = i + 8*hi, column pi
#pragma unroll
        for (int i = 0; i < 8; ++i) {
            bool lt = c[i] < minv[i];
            minv[i] = lt ? c[i] : minv[i];
            mini[i] = lt ? pi   : mini[i];
        }
    }

    // Cross-lane argmin over the 16 lanes of each half (xor masks stay
    // in-half); tie-break on lower index here for determinism.
#pragma unroll
    for (int i = 0; i < 8; ++i) {
        float v  = minv[i];
        int   ix = mini[i];
#pragma unroll
        for (int ofs = 1; ofs < 16; ofs <<= 1) {
            float ov = __shfl_xor(v,  ofs, 32);
            int   oi = __shfl_xor(ix, ofs, 32);
            if (ov < v || (ov == v && oi < ix)) { v = ov; ix = oi; }
        }
        mini[i] = ix;
    }

    if ((lane & 15) == 0) {
#pragma unroll
        for (int i = 0; i < 8; ++i)
            closest[s0 + hi * 8 + i] = mini[i];   // rows M = 8*hi + i
    }
}

// ---------------------------------------------------------------------------
// Kernel 2: quadric scatter (segment sums) via fp32 global atomics.
// ---------------------------------------------------------------------------
__global__ __launch_bounds__(256) void ponq_scatter(
    const float* __restrict__ samples,
    const float* __restrict__ normals,
    const int*   __restrict__ closest,
    float* __restrict__ Qacc,     // 2048 x 16
    float* __restrict__ nsum,     // 2048 x 3   (mean_normals slot)
    float* __restrict__ counts)   // 2048       (non_void slot)
{
    int i = blockIdx.x * blockDim.x + threadIdx.x;
    if (i >= NSAMP) return;
    float sx = samples[3 * i], sy = samples[3 * i + 1], sz = samples[3 * i + 2];
    float nx = normals[3 * i], ny = normals[3 * i + 1], nz = normals[3 * i + 2];
    float d  = -(nx * sx + ny * sy + nz * sz);
    float ps[4] = { nx, ny, nz, d };
    int c = closest[i];
    float* Qc = Qacc + c * 16;
#pragma unroll
    for (int r = 0; r < 4; ++r)
#pragma unroll
        for (int cc = 0; cc < 4; ++cc)
            atomicAdd(&Qc[r * 4 + cc], ps[r] * ps[cc]);
    atomicAdd(&nsum[3 * c + 0], nx);
    atomicAdd(&nsum[3 * c + 1], ny);
    atomicAdd(&nsum[3 * c + 2], nz);
    atomicAdd(&counts[c], 1.0f);
}

// ---------------------------------------------------------------------------
// Kernel 3: per-point 3x3 symmetric eigensolve (Jacobi) + pseudoinverse solve.
// ---------------------------------------------------------------------------
__device__ __forceinline__ void jrot(float A[3][3], float V[3][3], int p, int q, int r) {
    float apq = A[p][q];
    if (fabsf(apq) < 1e-30f) return;
    float app = A[p][p], aqq = A[q][q];
    float tau = (aqq - app) / (2.0f * apq);
    float t   = (tau >= 0.0f ? 1.0f : -1.0f) / (fabsf(tau) + sqrtf(1.0f + tau * tau));
    float cth = 1.0f / sqrtf(1.0f + t * t);
    float sth = t * cth;
    A[p][p] = app - t * apq;
    A[q][q] = aqq + t * apq;
    A[p][q] = 0.0f; A[q][p] = 0.0f;
    float arp = A[r][p], arq = A[r][q];
    A[r][p] = A[p][r] = cth * arp - sth * arq;
    A[r][q] = A[q][r] = sth * arp + cth * arq;
#pragma unroll
    for (int k = 0; k < 3; ++k) {
        float vkp = V[k][p], vkq = V[k][q];
        V[k][p] = cth * vkp - sth * vkq;
        V[k][q] = sth * vkp + cth * vkq;
    }
}

__global__ __launch_bounds__(256) void ponq_solve(
    const float* __restrict__ points,
    float* __restrict__ out)
{
    int i = blockIdx.x * blockDim.x + threadIdx.x;
    if (i >= NPTS) return;

    float* vstars = out;                        // [0,      6144)
    float* meann  = out + NPTS * 3;             // [6144,  12288)
    float* Q      = out + NPTS * 6;             // [12288, 45056)
    float* nvf    = out + NPTS * 6 + NPTS * 16; // [45056, 47104)

    float px = points[3 * i], py = points[3 * i + 1], pz = points[3 * i + 2];
    float cnt = nvf[i];

    float inv = 1.0f / fmaxf(cnt, 1.0f);
    meann[3 * i + 0] *= inv;
    meann[3 * i + 1] *= inv;
    meann[3 * i + 2] *= inv;

    bool non_void = cnt > 0.0f;
    nvf[i] = non_void ? 1.0f : 0.0f;

    if (!non_void) {
        vstars[3 * i + 0] = px;
        vstars[3 * i + 1] = py;
        vstars[3 * i + 2] = pz;
        return;
    }

    const float* Qi = Q + i * 16;
    float a00 = Qi[0], a01 = Qi[1], a02 = Qi[2];
    float a11 = Qi[5], a12 = Qi[6], a22 = Qi[10];
    float bx = -Qi[12], by = -Qi[13], bz = -Qi[14];

    float A[3][3] = { { a00, a01, a02 }, { a01, a11, a12 }, { a02, a12, a22 } };
    float V[3][3] = { { 1, 0, 0 }, { 0, 1, 0 }, { 0, 0, 1 } };
#pragma unroll
    for (int sweep = 0; sweep < 6; ++sweep) {
        jrot(A, V, 0, 1, 2);
        jrot(A, V, 0, 2, 1);
        jrot(A, V, 1, 2, 0);
    }

    float eig[3] = { A[0][0], A[1][1], A[2][2] };
    float vec[3][3];
#pragma unroll
    for (int k = 0; k < 3; ++k)
#pragma unroll
        for (int j = 0; j < 3; ++j)
            vec[k][j] = V[j][k];

#define CSWAP(x, y)                                                         \
    if (eig[y] > eig[x]) {                                                  \
        float te = eig[x]; eig[x] = eig[y]; eig[y] = te;                    \
        for (int j = 0; j < 3; ++j) {                                       \
            float tv = vec[x][j]; vec[x][j] = vec[y][j]; vec[y][j] = tv;    \
        }                                                                   \
    }
    CSWAP(0, 1) CSWAP(1, 2) CSWAP(0, 1)
#undef CSWAP

    float emax = fmaxf(eig[0], 1e-12f);
    float invE[3];
#pragma unroll
    for (int k = 0; k < 3; ++k) {
        bool mask = (eig[k] / emax) > EPS_R;
        invE[k] = mask ? (1.0f / eig[k]) : 0.0f;
    }

    float rx = bx - (a00 * px + a01 * py + a02 * pz);
    float ry = by - (a01 * px + a11 * py + a12 * pz);
    float rz = bz - (a02 * px + a12 * py + a22 * pz);

    float cx = 0.0f, cy = 0.0f, cz = 0.0f;
#pragma unroll
    for (int k = 0; k < 3; ++k) {
        float tk = invE[k] * (vec[k][0] * rx + vec[k][1] * ry + vec[k][2] * rz);
        cx += vec[k][0] * tk;
        cy += vec[k][1] * tk;
        cz += vec[k][2] * tk;
    }

    vstars[3 * i + 0] = px + cx;
    vstars[3 * i + 1] = py + cy;
    vstars[3 * i + 2] = pz + cz;
}

// ---------------------------------------------------------------------------
extern "C" void kernel_launch(void* const* d_in, const int* in_sizes, int n_in,
                              void* d_out, int out_size, void* d_ws, size_t ws_size,
                              hipStream_t stream) {
    const float* points  = (const float*)d_in[0];
    const float* samples = (const float*)d_in[1];
    const float* normals = (const float*)d_in[2];
    float* out = (float*)d_out;
    int* closest = (int*)d_ws;   // 131072 ints

    const int total_out = NPTS * 3 + NPTS * 3 + NPTS * 16 + NPTS;  // 47104
    ponq_zero<<<(total_out + 255) / 256, 256, 0, stream>>>(out, total_out);

    ponq_closest<<<NSAMP / 128, 256, 0, stream>>>(points, samples, closest);

    ponq_scatter<<<NSAMP / 256, 256, 0, stream>>>(
        samples, normals, closest,
        out + NPTS * 6,            // Q accumulator (final Q slot)
        out + NPTS * 3,            // nsum -> mean_normals slot
        out + NPTS * 6 + NPTS * 16 // counts -> non_void slot
    );

    ponq_solve<<<(NPTS + 255) / 256, 256, 0, stream>>>(points, out);
}